// YOLOV5TorchObjectDetector_51445118271890
// MI455X (gfx1250) — compile-verified
//
#include <hip/hip_runtime.h>
#include <hip/hip_bf16.h>

// ---------------- problem constants (match reference) ----------------
constexpr int   BATCH    = 8;
constexpr int   NROWS    = 25200;
constexpr int   NGRP     = NROWS / 4;   // 6300 float4 groups
constexpr int   NCLS     = 80;
constexpr int   MAXDET   = 300;
constexpr float CONF_THR = 0.4f;
constexpr float IOU_THR  = 0.45f;
constexpr float MAX_WH   = 4096.0f;

constexpr int NPAD = 25216;   // NROWS padded to multiple of 16 for LDS layout
constexpr int NT   = 1024;    // threads per NMS block (32 wave32s)

// ---------------- gfx1250 async-to-LDS helpers (compile-verified round 1) ----------------
__device__ __forceinline__ void async_load_b128_to_lds(const float* g, float* lds) {
#if defined(__HIP_DEVICE_COMPILE__) && defined(__gfx1250__)
    // VDST = VGPR holding LDS byte address; VADDR = 64-bit global address.
    // Flat pointers into the LDS aperture carry the LDS offset in bits [31:0]
    // (CDNA5 ISA 10.2), so truncating the generic shared pointer is the DS address.
    unsigned laddr = (unsigned)(unsigned long long)lds;
    unsigned long long gaddr = (unsigned long long)g;
    asm volatile("global_load_async_to_lds_b128 %0, %1, off"
                 :: "v"(laddr), "v"(gaddr)
                 : "memory");
#else
    lds[0] = g[0]; lds[1] = g[1]; lds[2] = g[2]; lds[3] = g[3];
#endif
}

__device__ __forceinline__ void wait_async_zero() {
#if defined(__HIP_DEVICE_COMPILE__) && defined(__gfx1250__)
#if __has_builtin(__builtin_amdgcn_s_wait_asynccnt)
    __builtin_amdgcn_s_wait_asynccnt(0);
#else
    asm volatile("s_wait_asynccnt 0" ::: "memory");
#endif
#if __has_builtin(__builtin_amdgcn_s_wait_tensorcnt)
    __builtin_amdgcn_s_wait_tensorcnt(0);   // probe-confirmed gfx1250 wait path
#endif
#endif
}

// (max score, min index) combine — matches jnp.argmax first-occurrence tie-break
__device__ __forceinline__ void amax_combine(float& bv, int& bi, float ov, int oi) {
    if (ov > bv || (ov == bv && oi < bi)) { bv = ov; bi = oi; }
}

// ---------------- kernel 1: per-row prep (conf/cls/box/obox/score0) ----------------
__global__ void yolo_prep_kernel(const float* __restrict__ pred,
                                 float* __restrict__ score,
                                 float* __restrict__ conf_o,
                                 float* __restrict__ cls_o,
                                 float4* __restrict__ box_o,
                                 float4* __restrict__ obox_o) {
    int idx = blockIdx.x * blockDim.x + threadIdx.x;
    if (idx >= BATCH * NROWS) return;
    const float* p = pred + (size_t)idx * (5 + NCLS);

    float cx = p[0], cy = p[1], w = p[2], h = p[3], obj = p[4];

    // conf = max_c(cls_c * obj), cls = first argmax (strict > keeps first occurrence)
    float best = p[5] * obj;
    int   bc   = 0;
#pragma unroll 8
    for (int c = 1; c < NCLS; ++c) {
        float s = p[5 + c] * obj;
        if (s > best) { best = s; bc = c; }
    }
    bool valid = (obj > CONF_THR) && (best > CONF_THR);

    float x1 = cx - w * 0.5f, y1 = cy - h * 0.5f;
    float x2 = cx + w * 0.5f, y2 = cy + h * 0.5f;
    float clsf = (float)bc;
    float off  = clsf * MAX_WH;

    score[idx]  = valid ? best : -1.0f;
    conf_o[idx] = best;
    cls_o[idx]  = clsf;
    box_o[idx]  = make_float4(x1, y1, x2, y2);
    obox_o[idx] = make_float4(x1 + off, y1 + off, x2 + off, y2 + off);
}

// ---------------- kernel 2: greedy NMS, one block per image ----------------
// Per iteration: ONE fused pass (suppress previous winner + local argmax),
// wave32 shuffle butterflies for reduction, only 2 barriers per step.
__global__ __launch_bounds__(NT)
void yolo_nms_kernel(const float* __restrict__ score_g,
                     const float* __restrict__ conf_g,
                     const float* __restrict__ cls_g,
                     const float4* __restrict__ box_g,
                     const float4* __restrict__ obox_g,
                     const float* __restrict__ logits,
                     float* __restrict__ out) {
    extern __shared__ __align__(16) float smem[];
    float*  s_score  = smem;                       // NPAD floats (mutable scores)
    float4* s_score4 = (float4*)smem;
    float*  s_rval   = smem + NPAD;                // 32 per-wave maxima
    int*    s_ridx   = (int*)(smem + NPAD + 32);   // 32 per-wave arg indices
    float*  s_q      = smem + NPAD + 64;           // 8 broadcast slots

    const int t    = threadIdx.x;
    const int lane = t & 31;
    const int wid  = t >> 5;
    const int b    = blockIdx.x;
    const size_t base = (size_t)b * NROWS;

    float* dets_b = out + (size_t)b * MAXDET * 6;
    float* logi_b = out + (size_t)BATCH * MAXDET * 6 + (size_t)b * MAXDET * NCLS;
    float* keep_b = out + (size_t)BATCH * MAXDET * (6 + NCLS) + (size_t)b * MAXDET;

    // ---- stage scores into LDS via gfx1250 async-to-LDS (B128 = 4 floats/op) ----
    for (int g = t; g < NGRP; g += NT)
        async_load_b128_to_lds(score_g + base + (size_t)g * 4, &s_score[g * 4]);
    wait_async_zero();
    __syncthreads();

    // previous-winner query box (uniform across block once broadcast)
    bool  have_prev = false;
    float qx1 = 0.f, qy1 = 0.f, qx2 = 0.f, qy2 = 0.f, qa = 0.f;

    for (int j = 0; j < MAXDET; ++j) {
        // ---- fused pass: suppress previous winner, fold into local argmax ----
        float bv = -3.4e38f;
        int   bi = 0x7FFFFFFF;
        for (int g = t; g < NGRP; g += NT) {
            float4 sv = s_score4[g];
            if (have_prev) {
                const float4* ob = obox_g + base + (size_t)g * 4;
                bool changed = false;
#pragma unroll
                for (int k = 0; k < 4; ++k) {
                    float4 o = ob[k];
                    float lx = fmaxf(qx1, o.x), ly = fmaxf(qy1, o.y);
                    float rx = fminf(qx2, o.z), ry = fminf(qy2, o.w);
                    float iw = fmaxf(rx - lx, 0.0f), ih = fmaxf(ry - ly, 0.0f);
                    float inter = iw * ih;
                    float a2  = (o.z - o.x) * (o.w - o.y);
                    float iou = inter / (qa + a2 - inter + 1e-9f);  // same assoc. as ref
                    if (iou > IOU_THR) {
                        (&sv.x)[k] = -1.0f;
                        changed = true;
                    }
                }
                if (changed) s_score4[g] = sv;
            }
            int ibase = g * 4;   // ascending index order preserves first-argmax
            if (sv.x > bv) { bv = sv.x; bi = ibase;     }
            if (sv.y > bv) { bv = sv.y; bi = ibase + 1; }
            if (sv.z > bv) { bv = sv.z; bi = ibase + 2; }
            if (sv.w > bv) { bv = sv.w; bi = ibase + 3; }
        }

        // ---- wave32 butterfly reduction (no barrier) ----
#pragma unroll
        for (int off = 16; off > 0; off >>= 1)
            amax_combine(bv, bi, __shfl_xor(bv, off, 32), __shfl_xor(bi, off, 32));
        if (lane == 0) { s_rval[wid] = bv; s_ridx[wid] = bi; }
        __syncthreads();

        // ---- wave 0 reduces the 32 per-wave partials, broadcasts winner ----
        if (t < 32) {
            bv = s_rval[t]; bi = s_ridx[t];
#pragma unroll
            for (int off = 16; off > 0; off >>= 1)
                amax_combine(bv, bi, __shfl_xor(bv, off, 32), __shfl_xor(bi, off, 32));
            if (t == 0) {
                int keep = (bv > 0.0f) ? 1 : 0;
                float4 q = obox_g[base + bi];
                s_q[0] = q.x; s_q[1] = q.y; s_q[2] = q.z; s_q[3] = q.w;
                s_q[4] = (q.z - q.x) * (q.w - q.y);   // a1
                s_q[5] = __int_as_float(bi);
                s_q[6] = __int_as_float(keep);
            }
        }
        __syncthreads();

        const int idx  = __float_as_int(s_q[5]);
        const int keep = __float_as_int(s_q[6]);

        if (!keep) {
            // scores never change again -> all remaining rows are zeros in the reference
            int rem = MAXDET - j;
            for (int k = t; k < rem * 6;    k += NT) dets_b[j * 6 + k]    = 0.0f;
            for (int k = t; k < rem * NCLS; k += NT) logi_b[j * NCLS + k] = 0.0f;
            for (int k = t; k < rem;        k += NT) keep_b[j + k]        = 0.0f;
            break;   // uniform branch (same keep in every lane)
        }

        // emit detection j (keep==1 -> multiplier is 1)
        if (t == 0) {
            float4 bx = box_g[base + idx];
            dets_b[j * 6 + 0] = bx.x;
            dets_b[j * 6 + 1] = bx.y;
            dets_b[j * 6 + 2] = bx.z;
            dets_b[j * 6 + 3] = bx.w;
            dets_b[j * 6 + 4] = conf_g[base + idx];
            dets_b[j * 6 + 5] = cls_g[base + idx];
            keep_b[j] = 1.0f;
        }
        if (t < NCLS)
            logi_b[j * NCLS + t] = logits[(base + idx) * NCLS + t];

        // carry winner's offset box into the next fused pass
        qx1 = s_q[0]; qy1 = s_q[1]; qx2 = s_q[2]; qy2 = s_q[3]; qa = s_q[4];
        have_prev = true;
    }
}

// ---------------- launcher ----------------
extern "C" void kernel_launch(void* const* d_in, const int* in_sizes, int n_in,
                              void* d_out, int out_size, void* d_ws, size_t ws_size,
                              hipStream_t stream) {
    const float* pred   = (const float*)d_in[0];   // (8, 25200, 85) f32
    const float* logits = (const float*)d_in[1];   // (8, 25200, 80) f32
    float*       out    = (float*)d_out;           // dets | logits | keeps (flat)
    float*       ws     = (float*)d_ws;

    const int total = BATCH * NROWS;               // 201600
    float*  score = ws;                            // [total]
    float*  conf  = ws + total;                    // [total]
    float*  cls   = ws + 2 * total;                // [total]
    float4* box   = (float4*)(ws + 3 * total);     // [total] float4 (16B aligned)
    float4* obox  = (float4*)(ws + 3 * total + 4 * total);  // [total] float4

    yolo_prep_kernel<<<(total + 255) / 256, 256, 0, stream>>>(
        pred, score, conf, cls, box, obox);

    size_t shmem = (size_t)(NPAD + 64 + 8) * sizeof(float);  // ~99 KB < 320 KB/WGP
    yolo_nms_kernel<<<BATCH, NT, shmem, stream>>>(
        score, conf, cls, box, obox, logits, out);
}